// TFLOPLoss_76098230551044
// MI455X (gfx1250) — compile-verified
//
#include <hip/hip_runtime.h>
#include <hip/hip_bf16.h>
#include <math.h>

typedef __attribute__((ext_vector_type(2))) float v2f;
typedef __attribute__((ext_vector_type(8))) float v8f;

#define B_   16
#define L_   688
#define C_   16
#define M_   4
#define NR   (B_ * L_)     // 11008
#define TWOL (2 * L_)      // 1376
#define LSTRIDE 132        // LDS row stride (floats); 528B per row keeps b64/b128 alignment + spreads banks

// ---------------- wave32 reductions ----------------
__device__ __forceinline__ float waveSum(float v) {
#pragma unroll
  for (int off = 16; off > 0; off >>= 1) v += __shfl_xor(v, off, 32);
  return v;
}
__device__ __forceinline__ float waveMax(float v) {
#pragma unroll
  for (int off = 16; off > 0; off >>= 1) v = fmaxf(v, __shfl_xor(v, off, 32));
  return v;
}

// ---------------- tag loss: class counts (deterministic int atomics) ----------------
__global__ void k_tag_counts(const int* __restrict__ tgt, float* __restrict__ ws) {
  __shared__ int cnts[C_];
  int tid = threadIdx.x;
  if (tid < C_) cnts[tid] = 0;
  __syncthreads();
  for (int i = tid; i < NR; i += 256) {
    int t = tgt[i];
    if (t != 0 && t != 1) atomicAdd(&cnts[t & (C_ - 1)], 1);
  }
  __syncthreads();
  if (tid == 0) {
    float total = 0.f;
    for (int c = 0; c < C_; ++c) { float f = (float)cnts[c]; ws[c] = f; total += f; }
    ws[16] = total;
  }
}

// ---------------- tag loss: per-row weighted NLL partials ----------------
__global__ void k_tag_rows(const float* __restrict__ lg, const int* __restrict__ tgt,
                           const float* __restrict__ ws,
                           float* __restrict__ pw, float* __restrict__ pwn) {
  int i = blockIdx.x * 256 + threadIdx.x;
  if (i >= NR) return;
  const float4* r4 = (const float4*)(lg + (size_t)i * C_);   // 64B-aligned row
  float4 q0 = r4[0], q1 = r4[1], q2 = r4[2], q3 = r4[3];
  float x[16] = {q0.x, q0.y, q0.z, q0.w, q1.x, q1.y, q1.z, q1.w,
                 q2.x, q2.y, q2.z, q2.w, q3.x, q3.y, q3.z, q3.w};
  float mx = x[0];
#pragma unroll
  for (int c = 1; c < C_; ++c) mx = fmaxf(mx, x[c]);
  float se = 0.f;
#pragma unroll
  for (int c = 0; c < C_; ++c) se += expf(x[c] - mx);
  float lse = mx + logf(se);
  int t = tgt[i];
  float nll = lse - x[t & (C_ - 1)];
  bool valid = (t != 0) && (t != 1);
  bool otsl = (t >= 2) && (t <= 6);
  float cn = ws[t & (C_ - 1)];
  float wgt = (otsl && cn > 0.f) ? (ws[16] / fmaxf(cn, 1.f)) : 1.f;
  float w = valid ? wgt : 0.f;
  pwn[i] = w * nll;
  pw[i]  = w;
}

// ---------------- pointer loss: WMMA GEMV denom (einsum bqk,bk->bq) ----------------
// Block = 256 thr = 8 waves handles 16 q-rows of one batch. exp(l/T)*dt staged to LDS via
// b128 loads/stores, row-sums computed on the matrix pipe with all-ones B (layout-invariant).
__global__ void k_ptr(const float* __restrict__ plog, const unsigned char* __restrict__ dtm,
                      const int* __restrict__ box,
                      float* __restrict__ ptot, float* __restrict__ pcnt) {
  __shared__ float tile[16 * LSTRIDE];
  __shared__ float partials[8][16];
  int tid = threadIdx.x;
  int b = blockIdx.x / 43, g = blockIdx.x % 43;
  int q0 = g * 16;
  const float* Lg = plog + ((size_t)(b * L_ + q0)) * L_;
  const unsigned char* dtb = dtm + (size_t)b * TWOL + L_;
  int wave = tid >> 5, lane = tid & 31, m = lane & 15, hi = lane >> 4;
  int c4 = (tid & 31) * 4;   // column within 128-wide chunk (16B aligned)
  int r0 = tid >> 5;         // rows r0 and r0+8

  v8f acc = {0.f, 0.f, 0.f, 0.f, 0.f, 0.f, 0.f, 0.f};
  v2f ones; ones.x = 1.0f; ones.y = 1.0f;

  for (int k0 = 0; k0 < L_; k0 += 128) {
    __syncthreads();  // protect tile from previous iteration's WMMA readers
    // ---- stage: two independent b128 loads + one dword of dt bytes per thread ----
    float4 a0, a1;
    unsigned int d4 = 0;
    if (k0 + c4 < L_) {
      a0 = *(const float4*)(Lg + (size_t)r0 * L_ + k0 + c4);
      a1 = *(const float4*)(Lg + (size_t)(r0 + 8) * L_ + k0 + c4);
      d4 = *(const unsigned int*)(dtb + k0 + c4);
    } else {
      a0 = make_float4(0.f, 0.f, 0.f, 0.f);
      a1 = a0;
    }
    float4 e0, e1;
    e0.x = expf(a0.x * 10.0f) * (float)((d4      ) & 0xffu);
    e0.y = expf(a0.y * 10.0f) * (float)((d4 >>  8) & 0xffu);
    e0.z = expf(a0.z * 10.0f) * (float)((d4 >> 16) & 0xffu);
    e0.w = expf(a0.w * 10.0f) * (float)((d4 >> 24) & 0xffu);
    e1.x = expf(a1.x * 10.0f) * (float)((d4      ) & 0xffu);
    e1.y = expf(a1.y * 10.0f) * (float)((d4 >>  8) & 0xffu);
    e1.z = expf(a1.z * 10.0f) * (float)((d4 >> 16) & 0xffu);
    e1.w = expf(a1.w * 10.0f) * (float)((d4 >> 24) & 0xffu);
    *(float4*)(&tile[r0 * LSTRIDE + c4])       = e0;
    *(float4*)(&tile[(r0 + 8) * LSTRIDE + c4]) = e1;
    __syncthreads();
    // ---- matrix-pipe row sums: each wave owns a K-slice of 16 columns ----
    int kb = wave * 16;
#pragma unroll
    for (int s = 0; s < 4; ++s) {
      int kk = kb + s * 4 + 2 * hi;  // ISA 16x4 f32 A layout: lane<16 -> K{0,1}; lane>=16 -> K{2,3}
      v2f a;
      a.x = tile[m * LSTRIDE + kk];
      a.y = tile[m * LSTRIDE + kk + 1];
      acc = __builtin_amdgcn_wmma_f32_16x16x4_f32(false, a, false, ones,
                                                  (short)0, acc, false, false);
    }
  }
  // D layout: vgpr r: lanes0-15 -> M=r (N=lane); lanes16-31 -> M=8+r (N=lane-16)
  if (lane == 0) {
#pragma unroll
    for (int r2 = 0; r2 < 8; ++r2) partials[wave][r2] = acc[r2];
  } else if (lane == 16) {
#pragma unroll
    for (int r2 = 0; r2 < 8; ++r2) partials[wave][8 + r2] = acc[r2];
  }
  __syncthreads();
  if (tid < 16) {
    float denom = 0.f;
#pragma unroll
    for (int w = 0; w < 8; ++w) denom += partials[w][tid];  // fixed order -> deterministic
    float ld = logf(denom + 1e-10f);
    int q = q0 + tid;
    const int* bx = box + ((size_t)(b * L_ + q)) * M_;
    const float* rowp = plog + ((size_t)(b * L_ + q)) * L_;
    float tot = 0.f, c = 0.f;
    bool lead = true;
#pragma unroll
    for (int mm = 0; mm < M_; ++mm) {
      int bi = bx[mm];
      lead = lead && (bi != -1);
      int tg = bi < 0 ? 0 : (bi > L_ - 1 ? L_ - 1 : bi);
      if (lead && dtb[tg]) { tot += ld - rowp[tg] * 10.0f; c += 1.f; }  // -log(exp(x)/d)=log d - x
    }
    int idx = b * L_ + q;
    ptot[idx] = tot;
    pcnt[idx] = c;
  }
}

// ---------------- span contrastive: one wave per row, rows resident in VGPRs ----------------
__global__ void k_span(const float* __restrict__ sim, const float* __restrict__ coef,
                       float* __restrict__ pl, float* __restrict__ pv) {
  int wave = threadIdx.x >> 5;
  int lane = threadIdx.x & 31;
  int r = blockIdx.x * 8 + wave;  // 1376*8 == 11008 exactly
  int j = r % L_;                 // diagonal column
  const float* sp = sim + (size_t)r * L_;
  const float* cp = coef + (size_t)r * L_;
  // 688 = 5*128 + 48 -> 6 float4 iterations per lane (last partial)
  float sv[24], cv[24];
#pragma unroll
  for (int it = 0; it < 6; ++it) {
    int k = (lane + it * 32) * 4;
    float4 s4, c4;
    if (k < L_) {
      s4 = *(const float4*)(sp + k);
      c4 = *(const float4*)(cp + k);
    } else {
      s4 = make_float4(-3.0e38f, -3.0e38f, -3.0e38f, -3.0e38f);
      c4 = make_float4(-1.f, -1.f, -1.f, -1.f);
    }
    sv[4*it + 0] = s4.x * 10.0f; sv[4*it + 1] = s4.y * 10.0f;
    sv[4*it + 2] = s4.z * 10.0f; sv[4*it + 3] = s4.w * 10.0f;
    cv[4*it + 0] = c4.x; cv[4*it + 1] = c4.y; cv[4*it + 2] = c4.z; cv[4*it + 3] = c4.w;
  }
  float mx = -3.0e38f;
#pragma unroll
  for (int e = 0; e < 24; ++e) mx = fmaxf(mx, sv[e]);
  mx = waveMax(mx);

  float den = 0.f, scs = 0.f, scw = 0.f, fge = 0.f, fpos = 0.f;
#pragma unroll
  for (int it = 0; it < 6; ++it) {
#pragma unroll
    for (int u = 0; u < 4; ++u) {
      int e = 4 * it + u;
      int k = (lane + it * 32) * 4 + u;
      if (k < L_) {
        float ex = expf(sv[e] - mx);
        if (k != j) den += ex;              // denom = rowsum(exp) - diag
        float cvv = cv[e];
        if (cvv > 0.f) { scs += cvv * sv[e]; scw += cvv; fpos = 1.f; }
        if (cvv >= 0.f) fge = 1.f;
      }
    }
  }
  den  = waveSum(den);
  scs  = waveSum(scs);
  scw  = waveSum(scw);
  fge  = waveMax(fge);
  fpos = waveMax(fpos);
  if (lane == 0) {
    float ld = logf(den + 1e-10f);
    // sum pos*coef*logp = sum pos*coef*v - (m + log denom)*sum pos*coef
    float num = scs - (mx + ld) * scw;
    float lossj = -num / (scw + 1e-10f);
    bool valid = (fge > 0.f) && (fpos > 0.f);
    pl[r] = valid ? lossj : 0.f;
    pv[r] = valid ? 1.f : 0.f;
  }
}

// ---------------- empty-pointer BCE: one block per batch ----------------
__global__ void k_eptr(const float* __restrict__ x, const unsigned char* __restrict__ dtm,
                       const unsigned char* __restrict__ am, float* __restrict__ ws) {
  int b = blockIdx.x, tid = threadIdx.x;
  __shared__ float sS[256], sA[256];
  float aS = 0.f, aA = 0.f;
  for (int k = tid; k < L_; k += 256) {
    float xv = x[(size_t)b * L_ + k];
    bool att = am[(size_t)b * TWOL + L_ + k] != 0;
    bool dtb = dtm[(size_t)b * TWOL + L_ + k] != 0;
    float t = (att && !dtb) ? 1.f : 0.f;
    float bce = fmaxf(xv, 0.f) - xv * t + log1pf(expf(-fabsf(xv)));
    if (att) { aS += bce; aA += 1.f; }
  }
  sS[tid] = aS; sA[tid] = aA;
  __syncthreads();
  for (int s = 128; s > 0; s >>= 1) {
    if (tid < s) { sS[tid] += sS[tid + s]; sA[tid] += sA[tid + s]; }
    __syncthreads();
  }
  if (tid == 0) ws[21 + b] = sS[0] / fmaxf(sA[0], 1.f);
}

// ---------------- deterministic reductions: 4 pairs, one block each ----------------
__global__ void k_reduce_all(const float* __restrict__ P, float* __restrict__ ws) {
  __shared__ float sA[1024], sB[1024];
  int pair = blockIdx.x;                       // 0:tag 1:ptr 2:row 3:col
  const float* a = P + (size_t)(2 * pair) * NR;
  const float* b = P + (size_t)(2 * pair + 1) * NR;
  int outoff = (pair == 0) ? 17 : (pair == 1) ? 19 : (pair == 2) ? 37 : 39;
  int tid = threadIdx.x;
  float xa = 0.f, xb = 0.f;
  for (int i = tid; i < NR; i += 1024) { xa += a[i]; xb += b[i]; }
  sA[tid] = xa; sB[tid] = xb;
  __syncthreads();
  for (int s = 512; s > 0; s >>= 1) {
    if (tid < s) { sA[tid] += sA[tid + s]; sB[tid] += sB[tid + s]; }
    __syncthreads();
  }
  if (tid == 0) { ws[outoff] = sA[0]; ws[outoff + 1] = sB[0]; }
}

// ---------------- final combine ----------------
__global__ void k_combine(const float* __restrict__ ws, float* __restrict__ out) {
  float cls = ws[17] / fmaxf(ws[18], 1e-10f);
  float ptr = (ws[20] > 0.f) ? (ws[19] / fmaxf(ws[20], 1.f)) : 0.f;
  float ep = 0.f;
  for (int b = 0; b < B_; ++b) ep += ws[21 + b];
  ep *= (1.f / 16.f);
  float rowL = (ws[38] > 0.f) ? (ws[37] / fmaxf(ws[38], 1.f)) : 0.f;
  float colL = (ws[40] > 0.f) ? (ws[39] / fmaxf(ws[40], 1.f)) : 0.f;
  float total = cls + ptr + ep + 0.5f * rowL + 0.5f * colL;
  out[0] = total; out[1] = cls; out[2] = ptr; out[3] = ep; out[4] = rowL; out[5] = colL;
}

extern "C" void kernel_launch(void* const* d_in, const int* in_sizes, int n_in,
                              void* d_out, int out_size, void* d_ws, size_t ws_size,
                              hipStream_t stream) {
  (void)in_sizes; (void)n_in; (void)out_size; (void)ws_size;
  const float* tag_logits      = (const float*)d_in[0];
  const float* pointer_logits  = (const float*)d_in[1];
  const float* empty_ptr       = (const float*)d_in[2];
  const float* row_sim         = (const float*)d_in[3];
  const float* col_sim         = (const float*)d_in[4];
  const float* row_coef        = (const float*)d_in[5];
  const float* col_coef        = (const float*)d_in[6];
  const int*   tag_targets     = (const int*)d_in[7];
  const int*   box_indices     = (const int*)d_in[8];
  const unsigned char* dtm     = (const unsigned char*)d_in[9];   // bool (B, 2L)
  const unsigned char* am      = (const unsigned char*)d_in[10];  // bool (B, 2L)
  float* ws  = (float*)d_ws;
  float* out = (float*)d_out;

  // workspace: [0..15] counts, [16] total, [17..20] tag/ptr sums, [21..36] eptr per-b,
  // [37..40] row/col sums, [64..] partial arrays (8 x NR floats ~ 344 KB)
  float* P       = ws + 64;
  float* P_wnll  = P;               // pair 0
  float* P_w     = P + 1 * NR;
  float* P_ptot  = P + 2 * NR;      // pair 1
  float* P_pcnt  = P + 3 * NR;
  float* P_rl    = P + 4 * NR;      // pair 2
  float* P_rv    = P + 5 * NR;
  float* P_cl    = P + 6 * NR;      // pair 3
  float* P_cv    = P + 7 * NR;

  hipLaunchKernelGGL(k_tag_counts, dim3(1),    dim3(256),  0, stream, tag_targets, ws);
  hipLaunchKernelGGL(k_tag_rows,   dim3(43),   dim3(256),  0, stream, tag_logits, tag_targets, ws, P_w, P_wnll);
  hipLaunchKernelGGL(k_ptr,        dim3(688),  dim3(256),  0, stream, pointer_logits, dtm, box_indices, P_ptot, P_pcnt);
  hipLaunchKernelGGL(k_span,       dim3(1376), dim3(256),  0, stream, row_sim, row_coef, P_rl, P_rv);
  hipLaunchKernelGGL(k_span,       dim3(1376), dim3(256),  0, stream, col_sim, col_coef, P_cl, P_cv);
  hipLaunchKernelGGL(k_eptr,       dim3(16),   dim3(256),  0, stream, empty_ptr, dtm, am, ws);
  hipLaunchKernelGGL(k_reduce_all, dim3(4),    dim3(1024), 0, stream, P, ws);
  hipLaunchKernelGGL(k_combine,    dim3(1),    dim3(1),    0, stream, ws, out);
}